// BatchedLightSimulation_29618094473831
// MI455X (gfx1250) — compile-verified
//
#include <hip/hip_runtime.h>
#include <hip/hip_bf16.h>
#include <math.h>

// ---------------------------------------------------------------------------
// BatchedLightSimulation on gfx1250 (CDNA5), restructured as:
//   out[r, to] = gain[r%48] * sum_{t'} x[r, t'] * W[16*to - t']
// where W is the 16-pooled combination of the two causal conv kernels.
// Main kernel: V_WMMA_F32_16X16X4_F32 GEMM against a block-Toeplitz weight
// band staged in LDS.
// ---------------------------------------------------------------------------

typedef __attribute__((ext_vector_type(2))) float v2f;
typedef __attribute__((ext_vector_type(8))) float v8f;

#define NROW   1536          // 32 * 48
#define NTICK  6400
#define NOUT   400           // 6400 / 16
#define KTAPS  990
#define CLEN   1979          // 2*990 - 1
#define WLEN   1994          // taps of pooled kernel, index j -> W[j-15]
#define WPAD   256           // zero guard offset inside LDS copy
#define WLDS_N 2496          // covers idx in [0, 2480] with slack

// ---------------------------------------------------------------------------
// Kernel 0: build pooled weight array Wg[0..1993], Wg[j] = W[j-15].
// ---------------------------------------------------------------------------
__global__ __launch_bounds__(256) void light_prep_kernel(
    const float* __restrict__ p_logit,
    const float* __restrict__ p_log_tau_s,
    const float* __restrict__ p_log_tau_t,
    const float* __restrict__ p_period,
    const float* __restrict__ p_resp,
    float* __restrict__ Wg)
{
    __shared__ float sc[KTAPS];
    __shared__ float im[KTAPS];
    __shared__ float cc[CLEN];

    const float dt    = 0.1f;
    const float logit = p_logit[0];
    const float sf    = 1.0f / (1.0f + expf(-logit));
    const float tau_s = powf(10.0f, p_log_tau_s[0]);
    const float tau_t = powf(10.0f, p_log_tau_t[0]);
    const float Tp    = p_period[0];
    const float Tr    = p_resp[0];

    const float as = 1.0f - expf(-dt / tau_s);
    const float at = 1.0f - expf(-dt / tau_t);
    const float imps = (Tp * Tp + Tr * Tr) * dt / (Tp * Tr * Tr);

    for (int i = threadIdx.x; i < KTAPS; i += blockDim.x) {
        float t = (float)i * dt;
        sc[i] = sf * expf(-t / tau_s) * as + (1.0f - sf) * expf(-t / tau_t) * at;
        im[i] = expf(-t / Tr) * sinf(t / Tp) * imps;
    }
    __syncthreads();

    // cc = sc (x) im, full convolution, lags 0..1978
    for (int k = threadIdx.x; k < CLEN; k += blockDim.x) {
        int ilo = k - (KTAPS - 1); if (ilo < 0) ilo = 0;
        int ihi = k;               if (ihi > KTAPS - 1) ihi = KTAPS - 1;
        float s = 0.0f;
        for (int i = ilo; i <= ihi; ++i) s += sc[i] * im[k - i];
        cc[k] = s;
    }
    __syncthreads();

    // pooled kernel: W[m] = sum_{j=0..15} cc[m+j], m in [-15, 1978]
    for (int j = threadIdx.x; j < WLEN; j += blockDim.x) {
        int m = j - 15;
        float s = 0.0f;
        #pragma unroll
        for (int jj = 0; jj < 16; ++jj) {
            int idx = m + jj;
            if (idx >= 0 && idx < CLEN) s += cc[idx];
        }
        Wg[j] = s;
    }
}

// ---------------------------------------------------------------------------
// Kernel 1: banded Toeplitz GEMM with V_WMMA_F32_16X16X4_F32.
// grid = (25 tick-tiles, 24 row-macro-tiles), block = 128 (4 waves).
// Wave w computes the 16x16 output tile rows [(by*4+w)*16, +16) x ticks
// [bx*16, +16), accumulating over the valid K band in steps of 4.
// ---------------------------------------------------------------------------
__global__ __launch_bounds__(128) void light_conv_wmma_kernel(
    const float* __restrict__ x,     // (1536, 6400)
    const float* __restrict__ gain,  // (48,)
    const float* __restrict__ Wg,    // (1994,)
    float* __restrict__ out)         // (1536, 400)
{
    __shared__ float wl[WLDS_N];

    const int tid = threadIdx.x;
    // stage pooled kernel with zero guard bands: wl[WPAD + m] = W[m]
    for (int i = tid; i < WLDS_N; i += 128) {
        int j = i - (WPAD - 15);             // i - 241
        wl[i] = (j >= 0 && j < WLEN) ? Wg[j] : 0.0f;
    }
    __syncthreads();

    const int lane = tid & 31;
    const int wave = tid >> 5;
    const int T0   = blockIdx.x * 16;                 // output tick tile base
    const int r0   = (blockIdx.y * 4 + wave) * 16;    // row tile base

    const int nIdx = lane & 15;    // A: row within tile / B: col within tile / D: N
    const int hi   = lane >> 4;    // lane half
    const int k0   = hi * 2;       // K pair handled by this lane half

    // K band: need t'+k over [16*T0 - 1978, 16*T0 + 255] clipped to [0, 6399]
    int tlo = 16 * T0 - 1984; if (tlo < 0) tlo = 0;       // multiple of 16
    int thi = 16 * T0 + 252;  if (thi > NTICK - 4) thi = NTICK - 4;

    const float* arow = x + (size_t)(r0 + nIdx) * NTICK + k0;
    const int bbase = WPAD + 16 * (T0 + nIdx) - k0;

    v8f acc = {};
    #pragma unroll 2
    for (int t = tlo; t <= thi; t += 4) {
        // A fragment: 16x4 fp32, lane = M, (vgpr0, vgpr1) = (K=k0, K=k0+1)
        v2f a = *(const v2f*)(arow + t);
        // B fragment: 4x16 Toeplitz of W, lane = N, (vgpr0, vgpr1) = (K=k0, K=k0+1)
        int bi = bbase - t;
        v2f b;
        b.x = wl[bi];
        b.y = wl[bi - 1];
        acc = __builtin_amdgcn_wmma_f32_16x16x4_f32(
            /*neg_a=*/false, a, /*neg_b=*/false, b,
            /*c_mod=*/(short)0, acc, /*reuse_a=*/false, /*reuse_b=*/false);
    }

    // D: 16x16 fp32; vgpr v holds M = v (lanes 0-15) / v+8 (lanes 16-31), lane = N
    const int col = T0 + nIdx;
    #pragma unroll
    for (int v = 0; v < 8; ++v) {
        int row = r0 + v + 8 * hi;
        out[row * NOUT + col] = acc[v] * gain[row % 48];
    }
}

// ---------------------------------------------------------------------------
extern "C" void kernel_launch(void* const* d_in, const int* in_sizes, int n_in,
                              void* d_out, int out_size, void* d_ws, size_t ws_size,
                              hipStream_t stream)
{
    const float* timing   = (const float*)d_in[0];  // (32,48,6400) f32
    const float* sf_logit = (const float*)d_in[1];
    const float* lts      = (const float*)d_in[2];
    const float* ltt      = (const float*)d_in[3];
    const float* period   = (const float*)d_in[4];
    const float* resp     = (const float*)d_in[5];
    const float* gain     = (const float*)d_in[6];  // (48,) f32
    float* Wg  = (float*)d_ws;                      // 1994 floats of scratch
    float* out = (float*)d_out;                     // (32,48,400) f32

    light_prep_kernel<<<1, 256, 0, stream>>>(sf_logit, lts, ltt, period, resp, Wg);

    dim3 grid(NOUT / 16, NROW / 64);                // (25, 24)
    light_conv_wmma_kernel<<<grid, 128, 0, stream>>>(timing, gain, Wg, out);
}